// Model_86586540687786
// MI455X (gfx1250) — compile-verified
//
#include <hip/hip_runtime.h>
#include <cstdint>

// Problem constants (fixed by reference): B=256, D=8192, S=8, W=4, STATE_LEN=3, NSLOTS=2048
#define BB      256
#define DD      8192
#define SSQ     8
#define SL      3
#define NSLOTS_ 2048

// ---------------------------------------------------------------------------
// Kernel 1: bulk copy conv_states -> new_conv_states (201 MB) using the
// CDNA5 asynchronous global<->LDS data-mover path.
//   global_load_async_to_lds_b128  : 16 B/lane, memory -> LDS, ASYNCcnt
//   global_store_async_from_lds_b128: 16 B/lane, LDS -> memory, ASYNCcnt
// 4 chunks in flight per thread between s_wait_asynccnt fences (2 KB / wave).
// Total chunks = 2048*8192*3*4/16 = 12,582,912 = 2048 blocks * 256 thr * 24.
// ---------------------------------------------------------------------------
__global__ void __launch_bounds__(256)
copy_states_async(const char* __restrict__ src, char* __restrict__ dst) {
    __shared__ char smem[4 * 256 * 16];              // 16 KB staging
    const long long stride = 2048LL * 256LL * 16LL;  // bytes advanced per chunk-step
    const long long base =
        ((long long)blockIdx.x * 256 + threadIdx.x) * 16;

    uint32_t lds[4];
#pragma unroll
    for (int j = 0; j < 4; ++j)
        lds[j] = (uint32_t)(uintptr_t)(smem + (j * 256 + threadIdx.x) * 16);

    for (int t = 0; t < 6; ++t) {                    // 6 * 4 = 24 chunks/thread
#pragma unroll
        for (int j = 0; j < 4; ++j) {
            unsigned long long ga =
                (unsigned long long)(uintptr_t)(src + base + (long long)(t * 4 + j) * stride);
            asm volatile("global_load_async_to_lds_b128 %0, %1, off"
                         :: "v"(lds[j]), "v"(ga) : "memory");
        }
        asm volatile("s_wait_asynccnt 0x0" ::: "memory");   // loads landed in LDS
#pragma unroll
        for (int j = 0; j < 4; ++j) {
            unsigned long long gd =
                (unsigned long long)(uintptr_t)(dst + base + (long long)(t * 4 + j) * stride);
            asm volatile("global_store_async_from_lds_b128 %0, %1, off"
                         :: "v"(gd), "v"(lds[j]) : "memory");
        }
        asm volatile("s_wait_asynccnt 0x0" ::: "memory");   // LDS free for reuse (WAR)
    }
}

// ---------------------------------------------------------------------------
// Kernel 2: fused depthwise conv + residual + state writeback.
// One thread per (b,d). All global traffic is b128 except the 3-float state
// row (gather/scatter, 12 B contiguous per lane, lanes contiguous in d).
// Runs after the copy kernel on the same stream, so its state writes
// correctly overwrite the copied rows.
// ---------------------------------------------------------------------------
__global__ void __launch_bounds__(256)
conv_update(const float* __restrict__ x, const float* __restrict__ w,
            const float* __restrict__ cs, const long long* __restrict__ cidx,
            const int* __restrict__ res_p, const int* __restrict__ pad_p,
            float* __restrict__ out, float* __restrict__ st_out) {
    const int d = blockIdx.x * 256 + threadIdx.x;
    const int b = blockIdx.y;

    const long long idx   = cidx[b];
    const int       pad   = *pad_p;
    const int       res   = *res_p;
    const bool      valid = (idx != (long long)pad);
    const long long safe  = valid ? idx : 0;

    // gathered previous state (3 floats)
    const float* srow = cs + (safe * DD + d) * SL;
    const float  c0 = srow[0], c1 = srow[1], c2 = srow[2];

    // x[b,d,0..7] : two b128 loads
    const float4* xp = (const float4*)(x + ((long long)b * DD + d) * SSQ);
    const float4  xa = xp[0];
    const float4  xb = xp[1];

    // weight[d,0..3] : one b128 load
    const float4  wv = *(const float4*)(w + (long long)d * 4);

    const float cat[11] = { c0, c1, c2,
                            xa.x, xa.y, xa.z, xa.w,
                            xb.x, xb.y, xb.z, xb.w };

    float o[8];
#pragma unroll
    for (int s = 0; s < 8; ++s) {
        float acc = fmaf(cat[s],     wv.x,
                    fmaf(cat[s + 1], wv.y,
                    fmaf(cat[s + 2], wv.z,
                         cat[s + 3] * wv.w)));
        if (res) acc += cat[s + 3];          // residual: x[b,d,s] == cat[s+3]
        o[s] = acc;
    }

    float4* op = (float4*)(out + ((long long)b * DD + d) * SSQ);
    op[0] = make_float4(o[0], o[1], o[2], o[3]);
    op[1] = make_float4(o[4], o[5], o[6], o[7]);

    if (valid) {                             // new state = x[b,d,5:8]
        float* so = st_out + (idx * DD + d) * SL;
        so[0] = xb.y; so[1] = xb.z; so[2] = xb.w;
    }
}

// ---------------------------------------------------------------------------
// d_in order: x (f32, B*D*S), weight (f32, D*W), conv_states (f32, NSLOTS*D*SL),
//             cache_indices (i64, B), residual_connection (i32, 1), pad_slot_id (i32, 1)
// d_out: out (B*D*S floats) || new_conv_states (NSLOTS*D*SL floats)
// ---------------------------------------------------------------------------
extern "C" void kernel_launch(void* const* d_in, const int* in_sizes, int n_in,
                              void* d_out, int out_size, void* d_ws, size_t ws_size,
                              hipStream_t stream) {
    (void)in_sizes; (void)n_in; (void)out_size; (void)d_ws; (void)ws_size;

    const float*     x   = (const float*)d_in[0];
    const float*     w   = (const float*)d_in[1];
    const float*     cs  = (const float*)d_in[2];
    const long long* ci  = (const long long*)d_in[3];
    const int*       res = (const int*)d_in[4];
    const int*       pad = (const int*)d_in[5];

    float* out    = (float*)d_out;
    float* st_out = out + (long long)BB * DD * SSQ;   // 16,777,216 floats in

    // 1) stream full conv_states into the new_conv_states output region
    copy_states_async<<<2048, 256, 0, stream>>>((const char*)cs, (char*)st_out);

    // 2) fused conv + residual + per-slot state overwrite
    dim3 grid(DD / 256, BB);
    conv_update<<<grid, 256, 0, stream>>>(x, w, cs, ci, res, pad, out, st_out);
}